// SecondaryInputLayer_77318001263174
// MI455X (gfx1250) — compile-verified
//
#include <hip/hip_runtime.h>

// out[b,u] = inputs[b,u] * w[u,u]   (inputs @ diag(w))
// Memory-bound: ~268 MB total traffic -> HBM roofline ~11.5us on MI455X.
// Steady state per thread: NT b128 load -> mul -> NT b128 store, diag in reg.

typedef float v4f __attribute__((ext_vector_type(4)));
typedef int v4i __attribute__((ext_vector_type(4)));

#define UNITS 4096
#define VEC_PER_ROW (UNITS / 4) // 1024 float4 per row
#define BLOCK 256
#define GRID 1024
#define STRIDE ((long long)GRID * BLOCK) // 262144, multiple of VEC_PER_ROW

// ---- Kernel A: gather diag(w) into contiguous scratch (16 KB) ----
__global__ void diag_extract_kernel(const float* __restrict__ w,
                                    float* __restrict__ diag) {
    int i = blockIdx.x * blockDim.x + threadIdx.x;
    if (i < UNITS) {
        diag[i] = w[(long long)i * (UNITS + 1)];
    }
}

// ---- Kernel B: stream-scale columns by register-resident diag ----
// Requires gridDim.x == GRID, blockDim.x == BLOCK so that
// (tid + k*STRIDE) & (VEC_PER_ROW-1) is loop-invariant.
__global__ __launch_bounds__(BLOCK) void diag_scale_kernel(
    const float* __restrict__ in, const float* __restrict__ diag_src,
    int diag_elem_stride, float* __restrict__ out, long long total4) {
    __shared__ v4f sdiag4[BLOCK]; // this block's 4 KB diag slice

    const int t = threadIdx.x;
    const long long tid = (long long)blockIdx.x * BLOCK + t;
    const int c4 = (int)(tid & (VEC_PER_ROW - 1)); // invariant column group

    v4f d;
    if (diag_elem_stride == 1) {
        // Stage this block's 256-vec4 diag slice via gfx1250 async
        // global->LDS DMA (ASYNCcnt-tracked), one b128 per lane.
        const int slice_base = (int)(((long long)blockIdx.x * BLOCK) &
                                     (VEC_PER_ROW - 1)); // == c4 of lane 0
        const v4f* g4 = (const v4f*)diag_src;
#if __has_builtin(__builtin_amdgcn_global_load_async_to_lds_b128)
        __builtin_amdgcn_global_load_async_to_lds_b128(
            (__attribute__((address_space(1))) v4i*)(g4 + slice_base + t),
            (__attribute__((address_space(3))) v4i*)(sdiag4 + t),
            /*offset=*/0, /*cpol=*/0);
#if __has_builtin(__builtin_amdgcn_s_wait_asynccnt)
        __builtin_amdgcn_s_wait_asynccnt(0);
#else
        asm volatile("s_wait_asynccnt 0" ::: "memory");
#endif
#else
        sdiag4[t] = g4[slice_base + t];
#endif
        __syncthreads();
        d = sdiag4[t];
    } else {
        // Workspace-less fallback: gather 4 strided diag elems from w.
        const long long base = (long long)c4 * 4;
        d.x = diag_src[(base + 0) * diag_elem_stride];
        d.y = diag_src[(base + 1) * diag_elem_stride];
        d.z = diag_src[(base + 2) * diag_elem_stride];
        d.w = diag_src[(base + 3) * diag_elem_stride];
    }

    const v4f* __restrict__ in4 = (const v4f*)in;
    v4f* __restrict__ out4 = (v4f*)out;

    long long i = tid;
    // 4-way unrolled pure streaming loop: independent NT b128 loads for
    // memory-level parallelism; NT stores keep the one-shot stream out of L2.
    for (; i + 3 * STRIDE < total4; i += 4 * STRIDE) {
        v4f x0 = __builtin_nontemporal_load(&in4[i]);
        v4f x1 = __builtin_nontemporal_load(&in4[i + STRIDE]);
        v4f x2 = __builtin_nontemporal_load(&in4[i + 2 * STRIDE]);
        v4f x3 = __builtin_nontemporal_load(&in4[i + 3 * STRIDE]);
        __builtin_nontemporal_store(x0 * d, &out4[i]);
        __builtin_nontemporal_store(x1 * d, &out4[i + STRIDE]);
        __builtin_nontemporal_store(x2 * d, &out4[i + 2 * STRIDE]);
        __builtin_nontemporal_store(x3 * d, &out4[i + 3 * STRIDE]);
    }
    for (; i < total4; i += STRIDE) {
        v4f x = __builtin_nontemporal_load(&in4[i]);
        __builtin_nontemporal_store(x * d, &out4[i]);
    }
}

extern "C" void kernel_launch(void* const* d_in, const int* in_sizes, int n_in,
                              void* d_out, int out_size, void* d_ws,
                              size_t ws_size, hipStream_t stream) {
    const float* inputs = (const float*)d_in[0]; // [BATCH, UNITS] fp32
    const float* w = (const float*)d_in[1];      // [UNITS, UNITS] fp32
    float* out = (float*)d_out;                  // [BATCH, UNITS] fp32

    const long long total4 = (long long)out_size / 4;

    if (ws_size >= (size_t)UNITS * sizeof(float)) {
        float* diag = (float*)d_ws;
        diag_extract_kernel<<<(UNITS + BLOCK - 1) / BLOCK, BLOCK, 0, stream>>>(
            w, diag);
        diag_scale_kernel<<<GRID, BLOCK, 0, stream>>>(inputs, diag, 1, out,
                                                      total4);
    } else {
        // No scratch: gather diag directly from w's strided diagonal.
        diag_scale_kernel<<<GRID, BLOCK, 0, stream>>>(inputs, w, UNITS + 1,
                                                      out, total4);
    }
}